// CapsuleLayer_47029891891748
// MI455X (gfx1250) — compile-verified
//
#include <hip/hip_runtime.h>
#include <hip/hip_bf16.h>

// Problem constants (from reference setup_inputs).
#define B_   32
#define I_   2048
#define AIN_ 16
#define D_   32
#define A_   16
#define DA_  512   // D_*A_
#define NUM_ROUTING_ 3

typedef float v2f __attribute__((ext_vector_type(2)));
typedef float v8f __attribute__((ext_vector_type(8)));

// ---------------------------------------------------------------------------
// Kernel 1: votes[b,i,o] = sum_a x[b,i,a] * w[i,a,o]
// Per capsule i: X_i (32x16) @ W_i (16x512) using V_WMMA_F32_16X16X4_F32.
// One workgroup (8 waves) per i. Wave w computes N-tiles [4w..4w+3] x 2 M-tiles.
// ---------------------------------------------------------------------------
__global__ void __launch_bounds__(256)
votes_wmma_kernel(const float* __restrict__ x,     // [B_, I_, AIN_]
                  const float* __restrict__ w,     // [I_, AIN_, DA_]
                  float* __restrict__ votes) {     // [B_, I_, DA_]
  const int i    = blockIdx.x;
  const int wave = threadIdx.x >> 5;   // 0..7
  const int lane = threadIdx.x & 31;
  const int lrow = lane & 15;          // 0..15
  const int lhi  = lane >> 4;          // 0 or 1

  const float* __restrict__ Xi = x + (size_t)i * AIN_;          // Xi[b*I_*16 + a]
  const float* __restrict__ Wi = w + (size_t)i * AIN_ * DA_;    // Wi[k*512 + o]

  // Hoist A fragments (reused across all 4 N-tiles of this wave).
  // ISA 32-bit A 16x4 layout: lanes 0-15 hold K={0,1}, lanes 16-31 hold K={2,3}.
  v2f afrag[2][4];
#pragma unroll
  for (int mt = 0; mt < 2; ++mt) {
    const size_t rowoff = (size_t)(mt * 16 + lrow) * (I_ * AIN_);
#pragma unroll
    for (int kk = 0; kk < 4; ++kk) {
      const int kbase = kk * 4 + lhi * 2;
      afrag[mt][kk].x = Xi[rowoff + kbase];
      afrag[mt][kk].y = Xi[rowoff + kbase + 1];
    }
  }

  for (int nt = 0; nt < 4; ++nt) {
    const int n0 = (wave * 4 + nt) * 16;
    // Hoist B fragments for this N-tile (reused across both M-tiles).
    // B 4x16: N across lanes 0-15, lanes 16-31 hold K={2,3} rows.
    v2f bfrag[4];
#pragma unroll
    for (int kk = 0; kk < 4; ++kk) {
      const int kbase = kk * 4 + lhi * 2;
      const int col = n0 + lrow;
      bfrag[kk].x = Wi[(size_t)kbase * DA_ + col];
      bfrag[kk].y = Wi[(size_t)(kbase + 1) * DA_ + col];
    }
#pragma unroll
    for (int mt = 0; mt < 2; ++mt) {
      v8f acc = {0.f, 0.f, 0.f, 0.f, 0.f, 0.f, 0.f, 0.f};
#pragma unroll
      for (int kk = 0; kk < 4; ++kk) {
        acc = __builtin_amdgcn_wmma_f32_16x16x4_f32(
            /*neg_a=*/false, afrag[mt][kk],
            /*neg_b=*/false, bfrag[kk],
            /*c_mod=*/(short)0, acc,
            /*reuse_a=*/false, /*reuse_b=*/false);
      }
      // C/D layout: lane L, VGPR r -> D[M = r + 8*(L/16)][N = L%16]
      const float* dummy = nullptr; (void)dummy;
#pragma unroll
      for (int r = 0; r < 8; ++r) {
        const int bb = mt * 16 + r + 8 * lhi;          // batch row
        votes[(size_t)bb * (I_ * DA_) + (size_t)i * DA_ + n0 + lrow] = acc[r];
      }
    }
  }
}

// ---------------------------------------------------------------------------
// Kernel 2: zero-init logits.
// ---------------------------------------------------------------------------
__global__ void zero_kernel(float* __restrict__ p, int n) {
  int idx = blockIdx.x * blockDim.x + threadIdx.x;
  if (idx < n) p[idx] = 0.f;
}

// ---------------------------------------------------------------------------
// Kernel 3: route[b,i,d] = softmax_d(logits[b,i,:]). One wave32 per row of 32.
// ---------------------------------------------------------------------------
__global__ void __launch_bounds__(256)
route_softmax_kernel(const float* __restrict__ logits,  // [B_*I_, 32]
                     float* __restrict__ route) {       // [B_*I_, 32]
  const int row = blockIdx.x * 8 + (threadIdx.x >> 5);  // 8 waves per block
  const int d   = threadIdx.x & 31;
  float v = logits[(size_t)row * D_ + d];
  float m = v;
#pragma unroll
  for (int off = 16; off > 0; off >>= 1)
    m = fmaxf(m, __shfl_xor(m, off, 32));
  float e = __expf(v - m);
  float s = e;
#pragma unroll
  for (int off = 16; off > 0; off >>= 1)
    s += __shfl_xor(s, off, 32);
  route[(size_t)row * D_ + d] = e / s;
}

// ---------------------------------------------------------------------------
// Kernel 4: preact[b,d,a] = sum_i route[b,i,d]*votes[b,i,d*16+a] + bias[d,a];
//           act = squash(preact) over a. One block per (b,d).
// ---------------------------------------------------------------------------
__global__ void __launch_bounds__(256)
preact_squash_kernel(const float* __restrict__ votes,   // [B_, I_, DA_]
                     const float* __restrict__ route,   // [B_, I_, D_]
                     const float* __restrict__ bias,    // [D_, A_]
                     float* __restrict__ act) {         // [B_, D_, A_]
  const int b = blockIdx.x >> 5;      // gridDim.x == B_*D_
  const int d = blockIdx.x & 31;
  const int a  = threadIdx.x & 15;    // atom
  const int ig = threadIdx.x >> 4;    // 0..15 i-groups

  const float* __restrict__ vb = votes + (size_t)b * (I_ * DA_) + d * A_ + a;
  const float* __restrict__ rb = route + (size_t)b * (I_ * D_) + d;

  float part = 0.f;
  for (int i = ig; i < I_; i += 16)
    part += rb[(size_t)i * D_] * vb[(size_t)i * DA_];

  __shared__ float sm[256];
  __shared__ float pre[16];
  __shared__ float n2sh;
  sm[threadIdx.x] = part;
  __syncthreads();
  if (threadIdx.x < 16) {
    float s = 0.f;
#pragma unroll
    for (int g = 0; g < 16; ++g) s += sm[g * 16 + threadIdx.x];
    s += bias[d * A_ + threadIdx.x];
    pre[threadIdx.x] = s;
  }
  __syncthreads();
  if (threadIdx.x == 0) {
    float n2 = 0.f;
#pragma unroll
    for (int aa = 0; aa < 16; ++aa) n2 += pre[aa] * pre[aa];
    n2sh = n2;
  }
  __syncthreads();
  if (threadIdx.x < 16) {
    const float n2 = n2sh;
    const float scale = __fsqrt_rn(n2) / (1.f + n2);   // t/||t|| * n2/(1+n2)
    act[((size_t)b * D_ + d) * A_ + threadIdx.x] = pre[threadIdx.x] * scale;
  }
}

// ---------------------------------------------------------------------------
// Kernel 5: logits[b,i,d] += sum_a votes[b,i,d*16+a] * act[b,d,a]
// One thread per (b,i,d); 64B contiguous vote read per thread.
// ---------------------------------------------------------------------------
__global__ void __launch_bounds__(256)
logits_update_kernel(const float* __restrict__ votes,   // [B_, I_, DA_]
                     const float* __restrict__ act,     // [B_, D_, A_]
                     float* __restrict__ logits) {      // [B_, I_, D_]
  const size_t idx = (size_t)blockIdx.x * blockDim.x + threadIdx.x; // B_*I_*D_
  const int d = (int)(idx & 31);
  const size_t bi = idx >> 5;           // b*I_ + i
  const int b = (int)(bi / I_);
  const float* __restrict__ v  = votes + bi * DA_ + d * A_;
  const float* __restrict__ ac = act + ((size_t)b * D_ + d) * A_;
  float s = 0.f;
#pragma unroll
  for (int a = 0; a < A_; ++a) s += v[a] * ac[a];
  logits[idx] += s;
}

// ---------------------------------------------------------------------------
extern "C" void kernel_launch(void* const* d_in, const int* in_sizes, int n_in,
                              void* d_out, int out_size, void* d_ws, size_t ws_size,
                              hipStream_t stream) {
  (void)in_sizes; (void)n_in; (void)out_size; (void)ws_size;
  const float* x    = (const float*)d_in[0];   // [B_, I_, AIN_, 1, 1]
  const float* w    = (const float*)d_in[1];   // [I_, AIN_, DA_, 1, 1]
  const float* bias = (const float*)d_in[2];   // [D_, A_, 1, 1]
  // d_in[3] = num_routing (==3, fixed at trace time)
  float* act = (float*)d_out;                  // [B_, D_, A_, 1, 1]

  // Workspace layout.
  float* votes  = (float*)d_ws;                          // 32*2048*512 f32 = 128 MB
  float* logits = votes + (size_t)B_ * I_ * DA_;         // 8 MB
  float* route  = logits + (size_t)B_ * I_ * D_;         // 8 MB

  // logits = 0
  {
    const int n = B_ * I_ * D_;
    zero_kernel<<<(n + 255) / 256, 256, 0, stream>>>(logits, n);
  }

  // votes = per-capsule GEMM (WMMA fp32)
  votes_wmma_kernel<<<I_, 256, 0, stream>>>(x, w, votes);

  // Dynamic routing (num_routing = 3).
  for (int t = 0; t < NUM_ROUTING_; ++t) {
    route_softmax_kernel<<<(B_ * I_) / 8, 256, 0, stream>>>(logits, route);
    preact_squash_kernel<<<B_ * D_, 256, 0, stream>>>(votes, route, bias, act);
    if (t < NUM_ROUTING_ - 1) {
      logits_update_kernel<<<(B_ * I_ * D_) / 256, 256, 0, stream>>>(votes, act, logits);
    }
  }
}